// KPConv_42666205119314
// MI455X (gfx1250) — compile-verified
//
#include <hip/hip_runtime.h>
#include <hip/hip_bf16.h>

typedef __attribute__((ext_vector_type(2))) float v2f;
typedef __attribute__((ext_vector_type(8))) float v8f;

// Problem constants (match reference)
constexpr int   NQ  = 50000;
constexpr int   MS  = 50000;
constexpr int   HN  = 32;
constexpr int   KK  = 15;     // real kernel points; padded to 16 for WMMA
constexpr int   CIN = 64;
constexpr int   CO  = 64;
constexpr float KPE = 1.2f;

constexpr int QT    = 16;     // queries per workgroup (WMMA M dim in phase B)
constexpr int NTHR  = 512;    // 16 waves

__global__ __launch_bounds__(NTHR)
void kpconv_wmma_f32(const float* __restrict__ q_pts,
                     const float* __restrict__ s_pts,
                     const int*   __restrict__ nb,
                     const float* __restrict__ x,
                     const float* __restrict__ kp,
                     const float* __restrict__ wt,
                     float*       __restrict__ out)
{
    // LDS: weighted[q][k(16)][c] 64KB, partial C tiles 16KB, staged neighbors ~8KB
    __shared__ float lds_weighted[QT * 16 * CIN];     // 64 KB
    __shared__ float lds_partial [16 * 16 * 16];      // 16 KB (one 16x16 tile per wave)
    __shared__ float lds_nc      [QT * HN * 3];       // centered neighbor coords
    __shared__ int   lds_idx     [QT * HN];
    __shared__ float lds_norm    [QT];

    const int tid  = threadIdx.x;
    const int lane = tid & 31;
    const int wave = tid >> 5;
    const int qblk = blockIdx.x * QT;

    // ---------------- Phase 0: gather neighbors, density count ----------------
    {
        const int q = wave;          // one wave per query, lane == neighbor h
        const int h = lane;
        const int n = qblk + q;
        const int idx = nb[n * HN + h];
        lds_idx[q * HN + h] = idx;
        float cx = 1e6f, cy = 1e6f, cz = 1e6f, fs = 0.0f;
        if (idx < MS) {
            cx = s_pts[idx * 3 + 0] - q_pts[n * 3 + 0];
            cy = s_pts[idx * 3 + 1] - q_pts[n * 3 + 1];
            cz = s_pts[idx * 3 + 2] - q_pts[n * 3 + 2];
            for (int c = 0; c < CIN; c += 4) {
                fs += x[idx * CIN + c + 0] + x[idx * CIN + c + 1]
                    + x[idx * CIN + c + 2] + x[idx * CIN + c + 3];
            }
        }
        lds_nc[(q * HN + h) * 3 + 0] = cx;
        lds_nc[(q * HN + h) * 3 + 1] = cy;
        lds_nc[(q * HN + h) * 3 + 2] = cz;
        const unsigned long long m = __ballot(fs > 0.0f);
        if (lane == 0) {
            const int cnt = (int)__popcll(m);
            lds_norm[q] = (float)(cnt > 0 ? cnt : 1);
        }
    }
    __syncthreads();

    // ------- Phase A: per-query weighted[k,c] = sum_h W[k,h] * X[h,c] -------
    // A fragment: M=k (16 rows, row 15 zero), K=4 neighbors per step.
    // B fragment: K=4 neighbors, N=16 input channels (4 c-tiles).
    {
        const int q   = wave;
        const int k   = lane & 15;
        const int col = lane & 15;
        float kx = 0.0f, ky = 0.0f, kz = 0.0f;
        if (k < KK) { kx = kp[k * 3]; ky = kp[k * 3 + 1]; kz = kp[k * 3 + 2]; }

        v8f acc[4];
        #pragma unroll
        for (int ct = 0; ct < 4; ++ct) acc[ct] = (v8f){};

        const int hofs = (lane >> 4) * 2;   // lanes 0-15 hold K=+0/+1, 16-31 hold K=+2/+3
        for (int hb = 0; hb < HN; hb += 4) {
            const int h0 = hb + hofs;
            v2f A;
            #pragma unroll
            for (int s = 0; s < 2; ++s) {
                const float nx = lds_nc[(q * HN + h0 + s) * 3 + 0] - kx;
                const float ny = lds_nc[(q * HN + h0 + s) * 3 + 1] - ky;
                const float nz = lds_nc[(q * HN + h0 + s) * 3 + 2] - kz;
                float w = 1.0f - sqrtf(nx * nx + ny * ny + nz * nz) * (1.0f / KPE);
                w = fmaxf(w, 0.0f);
                if (k >= KK) w = 0.0f;      // pad row of A
                A[s] = w;
            }
            const int i0 = lds_idx[q * HN + h0];
            const int i1 = lds_idx[q * HN + h0 + 1];
            #pragma unroll
            for (int ct = 0; ct < 4; ++ct) {
                v2f B;
                B.x = (i0 < MS) ? x[i0 * CIN + ct * 16 + col] : 0.0f;
                B.y = (i1 < MS) ? x[i1 * CIN + ct * 16 + col] : 0.0f;
                acc[ct] = __builtin_amdgcn_wmma_f32_16x16x4_f32(
                    false, A, false, B, (short)0, acc[ct], false, false);
            }
        }

        // C layout: VGPR r -> M = r (lanes 0-15) or r+8 (lanes 16-31), N = lane&15
        const int mrow = (lane >> 4) * 8;
        #pragma unroll
        for (int ct = 0; ct < 4; ++ct)
            #pragma unroll
            for (int r = 0; r < 8; ++r)
                lds_weighted[q * 1024 + (r + mrow) * CIN + ct * 16 + col] = acc[ct][r];
    }
    __syncthreads();

    // ---- Phase B: out[16 n, 64 d] = weighted[16 n, 1024] @ Wt[1024, 64] ----
    // wave -> (d-tile, K-quarter); 64 WMMA K-steps of 4 each; partials to LDS.
    {
        const int dtile = wave & 3;
        const int kq    = wave >> 2;          // 0..3, each covers 256 of K=1024
        const int col   = lane & 15;
        const int kofs  = (lane >> 4) * 2;
        const int d     = dtile * 16 + col;
        const int nrow  = lane & 15;          // A row = local query

        v8f acc = (v8f){};
        for (int s = 0; s < 64; ++s) {
            const int kb = kq * 256 + s * 4 + kofs;   // even; pair is (kb, kb+1)
            v2f A, B;
            A.x = lds_weighted[nrow * 1024 + kb];
            A.y = lds_weighted[nrow * 1024 + kb + 1];
            const int kp0 = kb >> 6, c0 = kb & 63;          // kk = kpoint*64 + c
            const int kp1 = (kb + 1) >> 6, c1 = (kb + 1) & 63;
            B.x = (kp0 < KK) ? wt[kp0 * (CIN * CO) + c0 * CO + d] : 0.0f;
            B.y = (kp1 < KK) ? wt[kp1 * (CIN * CO) + c1 * CO + d] : 0.0f;
            acc = __builtin_amdgcn_wmma_f32_16x16x4_f32(
                false, A, false, B, (short)0, acc, false, false);
        }
        const int mrow = (lane >> 4) * 8;
        #pragma unroll
        for (int r = 0; r < 8; ++r)
            lds_partial[wave * 256 + (r + mrow) * 16 + col] = acc[r];
    }
    __syncthreads();

    // ------------- Phase C: reduce K-quarters, normalize, store -------------
    for (int e = tid; e < QT * CO; e += NTHR) {
        const int n  = e >> 6;
        const int d  = e & 63;
        const int dt = d >> 4;
        const int cc = d & 15;
        float s = 0.0f;
        #pragma unroll
        for (int kq = 0; kq < 4; ++kq)
            s += lds_partial[(kq * 4 + dt) * 256 + n * 16 + cc];
        out[(qblk + n) * CO + d] = s / lds_norm[n];
    }
}

extern "C" void kernel_launch(void* const* d_in, const int* in_sizes, int n_in,
                              void* d_out, int out_size, void* d_ws, size_t ws_size,
                              hipStream_t stream) {
    const float* q_pts = (const float*)d_in[0];   // [N,3]
    const float* s_pts = (const float*)d_in[1];   // [M,3]
    const int*   nb    = (const int*)  d_in[2];   // [N,H]
    const float* x     = (const float*)d_in[3];   // [M,CIN]
    const float* kp    = (const float*)d_in[4];   // [K,3]
    const float* wt    = (const float*)d_in[5];   // [K,CIN,COUT]
    float*       out   = (float*)d_out;           // [N,COUT]

    const int n = in_sizes[0] / 3;                // 50000
    const int nblk = (n + QT - 1) / QT;           // 3125 (exact)
    kpconv_wmma_f32<<<nblk, NTHR, 0, stream>>>(q_pts, s_pts, nb, x, kp, wt, out);
}